// DiffAttention_6914897347196
// MI455X (gfx1250) — compile-verified
//
#include <hip/hip_runtime.h>
#include <hip/hip_bf16.h>
#include <math.h>

#define N_NODES 100000
#define N_EDGES 1600000
#define DIM 64

typedef float v2f __attribute__((ext_vector_type(2)));
typedef float v8f __attribute__((ext_vector_type(8)));

// ---------------------------------------------------------------------------
// Kernel 1: node projection  P = H @ W^T  (N x 64), plus  a = P @ w_attn.
// One wave per 16-row strip; 4 column tiles (16 cols each) accumulated with
// V_WMMA_F32_16X16X4_F32 over K=64 in 16 steps of K=4.
//
// f32 WMMA operand layout (ISA 7.12.2):
//   A (16x4, MxK):  lanes 0-15 -> M=lane,   VGPR0=K0, VGPR1=K1
//                   lanes16-31 -> M=lane-16,VGPR0=K2, VGPR1=K3
//   B (4x16, KxN):  lanes 0-15 -> N=lane,   VGPR0=K0, VGPR1=K1
//                   lanes16-31 -> N=lane-16,VGPR0=K2, VGPR1=K3
//   C/D (16x16):    VGPR r: lanes0-15 -> (M=r,  N=lane),
//                           lanes16-31-> (M=r+8,N=lane-16)
// ---------------------------------------------------------------------------
__global__ __launch_bounds__(256)
void proj_kernel(const float* __restrict__ Hsrc, const float* __restrict__ Hdst,
                 const float* __restrict__ W,    const float* __restrict__ wat,
                 float* __restrict__ Psrc, float* __restrict__ Pdst,
                 float* __restrict__ Asrc, float* __restrict__ Adst,
                 int nTiles) {
  const int wave = (blockIdx.x * blockDim.x + threadIdx.x) >> 5;
  const int lane = threadIdx.x & 31;
  if (wave >= nTiles) return;                 // wave-uniform: EXEC all ones inside

  const float* __restrict__ H = blockIdx.y ? Hdst : Hsrc;
  float* __restrict__ P = blockIdx.y ? Pdst : Psrc;
  float* __restrict__ A = blockIdx.y ? Adst : Asrc;

  const int rowBase = wave * 16;
  const int m      = lane & 15;               // M for A / N for B / N for C
  const int khalf  = (lane >> 4) << 1;        // 0 (lanes 0-15) or 2 (lanes 16-31)

  v8f c0 = {}, c1 = {}, c2 = {}, c3 = {};

  #pragma unroll
  for (int k0 = 0; k0 < DIM; k0 += 4) {
    const int kk = k0 + khalf;
    // A tile: rows rowBase..rowBase+15 of H, cols kk,kk+1
    v2f a  = *(const v2f*)(H + (size_t)(rowBase + m) * DIM + kk);
    // B tiles: B[k][n] = W[n][k] (P = H @ W^T), 4 column groups of 16
    v2f b0 = *(const v2f*)(W + (size_t)(m +  0) * DIM + kk);
    v2f b1 = *(const v2f*)(W + (size_t)(m + 16) * DIM + kk);
    v2f b2 = *(const v2f*)(W + (size_t)(m + 32) * DIM + kk);
    v2f b3 = *(const v2f*)(W + (size_t)(m + 48) * DIM + kk);
    c0 = __builtin_amdgcn_wmma_f32_16x16x4_f32(false, a, false, b0, (short)0, c0, false, false);
    c1 = __builtin_amdgcn_wmma_f32_16x16x4_f32(false, a, false, b1, (short)0, c1, false, false);
    c2 = __builtin_amdgcn_wmma_f32_16x16x4_f32(false, a, false, b2, (short)0, c2, false, false);
    c3 = __builtin_amdgcn_wmma_f32_16x16x4_f32(false, a, false, b3, (short)0, c3, false, false);
  }

  const int rowOff = (lane >> 4) << 3;        // 0 or 8
  const float w0 = wat[m], w1 = wat[m + 16], w2 = wat[m + 32], w3 = wat[m + 48];

  #pragma unroll
  for (int r = 0; r < 8; ++r) {
    const int row = rowBase + r + rowOff;
    const float e0 = c0[r], e1 = c1[r], e2 = c2[r], e3 = c3[r];
    float* __restrict__ prow = P + (size_t)row * DIM;
    prow[m]      = e0;
    prow[m + 16] = e1;
    prow[m + 32] = e2;
    prow[m + 48] = e3;
    // attention scalar: dot of this output row with w_attn, reduced across
    // the 16 lanes that hold the row (xor shuffles stay inside each half)
    float t = e0 * w0 + e1 * w1 + e2 * w2 + e3 * w3;
    t += __shfl_xor(t, 1, 32);
    t += __shfl_xor(t, 2, 32);
    t += __shfl_xor(t, 4, 32);
    t += __shfl_xor(t, 8, 32);
    if (m == 0) A[row] = t;
  }
}

// ---------------------------------------------------------------------------
// Kernel 2: CSR row offsets from sorted dst (lower_bound per node).
// ---------------------------------------------------------------------------
__global__ __launch_bounds__(256)
void row_start_kernel(const int* __restrict__ dst, int nEdges, int nNodes,
                      int* __restrict__ rowStart) {
  const int n = blockIdx.x * blockDim.x + threadIdx.x;
  if (n > nNodes) return;
  int lo = 0, hi = nEdges;
  while (lo < hi) {
    const int mid = (lo + hi) >> 1;
    if (dst[mid] < n) lo = mid + 1; else hi = mid;
  }
  rowStart[n] = lo;                            // rowStart[nNodes] == nEdges
}

// ---------------------------------------------------------------------------
// Kernel 3: per-node segment softmax + weighted aggregation + ELU.
// One wave (32 lanes) per destination node.  tanh is bounded, so exp(e)
// without the max-shift is exactly the reference softmax.
//   out[n] = elu( P_dst[n] - (sum_j w_j * P_src[src_j]) / (sum_j w_j) )
// ---------------------------------------------------------------------------
__global__ __launch_bounds__(256)
void aggregate_kernel(const int* __restrict__ src,
                      const int* __restrict__ rowStart,
                      const float* __restrict__ Psrc, const float* __restrict__ Pdst,
                      const float* __restrict__ Asrc, const float* __restrict__ Adst,
                      float* __restrict__ out) {
  const int n    = (blockIdx.x * blockDim.x + threadIdx.x) >> 5;
  const int lane = threadIdx.x & 31;
  if (n >= N_NODES) return;

  const int s0 = rowStart[n];
  const int s1 = rowStart[n + 1];
  float* __restrict__ o = out + (size_t)n * DIM;
  if (s0 == s1) {                              // empty segment -> elu(0) = 0
    o[lane] = 0.f;
    o[lane + 32] = 0.f;
    return;
  }

  const float aD = Adst[n];
  float acc0 = 0.f, acc1 = 0.f, ssum = 0.f;

  for (int base = s0; base < s1; base += 32) {
    const int j = base + lane;
    float w = 0.f;
    int   si = 0;
    if (j < s1) {
      si = src[j];
      w  = expf(tanhf(aD - Asrc[si]));
    }
    ssum += w;
    const int cnt = min(32, s1 - base);
    for (int t = 0; t < cnt; ++t) {
      const float wt = __shfl(w, t, 32);
      const int   sb = __shfl(si, t, 32);
      const float* __restrict__ p = Psrc + (size_t)sb * DIM;
      acc0 = fmaf(wt, p[lane],      acc0);     // coalesced 128B
      acc1 = fmaf(wt, p[lane + 32], acc1);     // coalesced 128B
    }
  }

  #pragma unroll
  for (int msk = 16; msk; msk >>= 1) ssum += __shfl_xor(ssum, msk, 32);

  const float inv = 1.0f / ssum;
  const float h0 = Pdst[(size_t)n * DIM + lane]      - acc0 * inv;
  const float h1 = Pdst[(size_t)n * DIM + lane + 32] - acc1 * inv;
  o[lane]      = (h0 > 0.f) ? h0 : expm1f(h0); // elu, alpha = 1
  o[lane + 32] = (h1 > 0.f) ? h1 : expm1f(h1);
}

// ---------------------------------------------------------------------------
extern "C" void kernel_launch(void* const* d_in, const int* in_sizes, int n_in,
                              void* d_out, int out_size, void* d_ws, size_t ws_size,
                              hipStream_t stream) {
  (void)in_sizes; (void)n_in; (void)out_size; (void)ws_size;

  const float* h_src = (const float*)d_in[0];
  const float* h_dst = (const float*)d_in[1];
  const float* W_fc  = (const float*)d_in[2];
  const float* w_at  = (const float*)d_in[3];
  const int*   src   = (const int*)d_in[4];
  const int*   dst   = (const int*)d_in[5];
  float* out = (float*)d_out;

  // workspace carve-out (all freshly written before being read; no init needed)
  char*  ws  = (char*)d_ws;
  size_t off = 0;
  auto carve = [&](size_t bytes) {
    void* p = ws + off;
    off = (off + bytes + 255) & ~(size_t)255;
    return p;
  };
  float* Psrc     = (float*)carve(sizeof(float) * (size_t)N_NODES * DIM);
  float* Pdst     = (float*)carve(sizeof(float) * (size_t)N_NODES * DIM);
  float* Asrc     = (float*)carve(sizeof(float) * (size_t)N_NODES);
  float* Adst     = (float*)carve(sizeof(float) * (size_t)N_NODES);
  int*   rowStart = (int*)  carve(sizeof(int)   * (size_t)(N_NODES + 1));

  // 1) WMMA projections for both node tables (grid.y selects src/dst)
  const int nTiles = N_NODES / 16;             // 6250, exact
  dim3 pgrid((nTiles + 7) / 8, 2, 1);
  proj_kernel<<<pgrid, 256, 0, stream>>>(h_src, h_dst, W_fc, w_at,
                                         Psrc, Pdst, Asrc, Adst, nTiles);

  // 2) CSR offsets from sorted dst
  row_start_kernel<<<(N_NODES + 1 + 255) / 256, 256, 0, stream>>>(
      dst, N_EDGES, N_NODES, rowStart);

  // 3) segment softmax + aggregation + ELU (one wave per node)
  aggregate_kernel<<<((size_t)N_NODES * 32 + 255) / 256, 256, 0, stream>>>(
      src, rowStart, Psrc, Pdst, Asrc, Adst, out);
}